// GCNConv_35442070126638
// MI455X (gfx1250) — compile-verified
//
#include <hip/hip_runtime.h>

#define F       128
#define ROWS    64
#define WPAD    136   // padded W row stride (floats) -> conflict-free B reads
#define APAD    132   // padded agg row stride (floats) -> conflict-free A reads
#define THREADS 256   // 8 waves (wave32)

typedef __attribute__((ext_vector_type(2))) float v2f;
typedef __attribute__((ext_vector_type(8))) float v8f;

__global__ __launch_bounds__(THREADS) void gcn_fused_kernel(
    const float* __restrict__ X,        // [N, 128]
    const float* __restrict__ W,        // [128, 128]
    const float* __restrict__ deg,      // [N]  (d^{-1/2})
    const int*   __restrict__ rowp,     // [N+1]
    const int*   __restrict__ colidx,   // [E]
    float*       __restrict__ out,      // [N, 128]
    int n)
{
    extern __shared__ float smem[];
    float* sW = smem;                 // 128 * 136 floats
    float* sA = smem + F * WPAD;      // 64  * 132 floats

    const int tid  = threadIdx.x;
    const int wave = tid >> 5;        // 0..7
    const int lane = tid & 31;

    // ---- stage W into LDS (row-major, padded rows), fully coalesced ----
    for (int i = tid; i < F * (F / 4); i += THREADS) {
        const int r = i >> 5;             // 32 float4 per row
        const int c = (i & 31) << 2;
        const float4 w = *(const float4*)(W + r * F + c);
        float* dst = sW + r * WPAD + c;
        dst[0] = w.x; dst[1] = w.y; dst[2] = w.z; dst[3] = w.w;
    }

    // ---- aggregation: wave handles 8 rows; lane covers 4 features ----
    const int rowBase = blockIdx.x * ROWS;
    const int c4 = lane << 2;
    for (int i = 0; i < ROWS / 8; ++i) {
        const int lrow = wave * (ROWS / 8) + i;
        const int row  = rowBase + lrow;
        float ax = 0.f, ay = 0.f, az = 0.f, aw = 0.f;
        if (row < n) {
            const int   s  = rowp[row];
            const int   e  = rowp[row + 1];
            const float di = deg[row];
            for (int ed = s; ed < e; ++ed) {
                const int   j   = colidx[ed];            // wave-uniform
                const float nrm = di * deg[j];           // wave-uniform
                const float4 x = *(const float4*)(X + (long long)j * F + c4);
                ax += nrm * x.x; ay += nrm * x.y;
                az += nrm * x.z; aw += nrm * x.w;
            }
        }
        float* dst = sA + lrow * APAD + c4;
        dst[0] = ax; dst[1] = ay; dst[2] = az; dst[3] = aw;
    }
    __syncthreads();

    // ---- GEMM tile: out[64x128] = sA[64x128] @ sW[128x128] via WMMA f32 16x16x4 ----
    const int half = lane >> 4;       // 0: K even pair, 1: K odd pair (per ISA A/B layout)
    const int l16  = lane & 15;
    const int colBase = wave << 4;    // wave -> its 16-column tile
    for (int tr = 0; tr < ROWS / 16; ++tr) {
        v8f acc = {};
        const float* Arow = sA + (tr * 16 + l16) * APAD;
        const float* Bcol = sW + colBase + l16;
        #pragma unroll
        for (int k0 = 0; k0 < F; k0 += 4) {
            const int ka = k0 + 2 * half;
            v2f a; a.x = Arow[ka];          a.y = Arow[ka + 1];
            v2f b; b.x = Bcol[ka * WPAD];   b.y = Bcol[(ka + 1) * WPAD];
            acc = __builtin_amdgcn_wmma_f32_16x16x4_f32(
                false, a, false, b, (short)0, acc, false, false);
        }
        // D layout: VGPR r -> row (tr*16 + half*8 + r), col = colBase + l16
        const int col      = colBase + l16;
        const int nodeBase = rowBase + tr * 16 + half * 8;
        #pragma unroll
        for (int r = 0; r < 8; ++r) {
            const int node = nodeBase + r;
            if (node < n) out[(long long)node * F + col] = acc[r];
        }
    }
}

extern "C" void kernel_launch(void* const* d_in, const int* in_sizes, int n_in,
                              void* d_out, int out_size, void* d_ws, size_t ws_size,
                              hipStream_t stream) {
    const float* X      = (const float*)d_in[0];
    const float* W      = (const float*)d_in[1];
    const float* deg    = (const float*)d_in[2];
    const int*   rowp   = (const int*)d_in[3];
    const int*   colidx = (const int*)d_in[4];
    float*       out    = (float*)d_out;

    const int n = in_sizes[2];                        // degrees has N elements
    const int blocks = (n + ROWS - 1) / ROWS;
    const size_t lds_bytes = (size_t)(F * WPAD + ROWS * APAD) * sizeof(float); // 103,424 B

    gcn_fused_kernel<<<blocks, THREADS, lds_bytes, stream>>>(
        X, W, deg, rowp, colidx, out, n);
}